// GAT_27401891348551
// MI455X (gfx1250) — compile-verified
//
#include <hip/hip_runtime.h>
#include <hip/hip_bf16.h>
#include <math.h>

typedef float v2f __attribute__((ext_vector_type(2)));
typedef float v8f __attribute__((ext_vector_type(8)));

#define DD 128
#define NNODE 40000
#define NEDGE 640000

// ---------------------------------------------------------------------------
// GEMM: z = h @ W1, znb = h @ W2   (znb seeds the z_i + aggregation buffer)
// One block = 16 rows of h; 8 waves = 8 column tiles of 16.
// V_WMMA_F32_16X16X4_F32, K-loop of 32 steps, A operand shared by both WMMAs.
// A 16x4 f32 layout: lanes 0-15 hold K={0,1}, lanes 16-31 hold K={2,3}.
// C/D 16x16 f32 layout: vgpr r -> row r + 8*(lane>=16), col = lane%16.
// ---------------------------------------------------------------------------
__global__ __launch_bounds__(256)
void gat_gemm_wmma(const float* __restrict__ h,
                   const float* __restrict__ W1,
                   const float* __restrict__ W2,
                   float* __restrict__ z,
                   float* __restrict__ znb)
{
    const int lane  = threadIdx.x & 31;
    const int wavew = threadIdx.x >> 5;          // 0..7 -> column tile
    const int row0  = blockIdx.x << 4;           // 16 rows per block (2500 blocks)
    const int col0  = wavew << 4;
    const int mn    = lane & 15;                 // row (A) / col (B,C) index
    const int khalf = (lane >> 4) << 1;          // 0 or 2

    const float* hrow = h + (size_t)(row0 + mn) * DD;

    v8f accZ = {0.f, 0.f, 0.f, 0.f, 0.f, 0.f, 0.f, 0.f};
    v8f accI = {0.f, 0.f, 0.f, 0.f, 0.f, 0.f, 0.f, 0.f};

    for (int k0 = 0; k0 < DD; k0 += 4) {
        if (k0 + 16 < DD) __builtin_prefetch(&hrow[k0 + 16], 0, 1);
        v2f a;
        a.x = hrow[k0 + khalf];
        a.y = hrow[k0 + khalf + 1];
        v2f b1, b2;
        b1.x = W1[(size_t)(k0 + khalf)     * DD + col0 + mn];
        b1.y = W1[(size_t)(k0 + khalf + 1) * DD + col0 + mn];
        b2.x = W2[(size_t)(k0 + khalf)     * DD + col0 + mn];
        b2.y = W2[(size_t)(k0 + khalf + 1) * DD + col0 + mn];
        accZ = __builtin_amdgcn_wmma_f32_16x16x4_f32(false, a, false, b1,
                                                     (short)0, accZ, false, false);
        accI = __builtin_amdgcn_wmma_f32_16x16x4_f32(false, a, false, b2,
                                                     (short)0, accI, false, false);
    }

    const int rowBase = row0 + ((lane >> 4) << 3);
    #pragma unroll
    for (int r = 0; r < 8; ++r) {
        size_t idx = (size_t)(rowBase + r) * DD + col0 + mn;
        z[idx]   = accZ[r];
        znb[idx] = accI[r];
    }
}

// ---------------------------------------------------------------------------
// Per-node attention contributions: a_src[n]=z[n].Wa[0:128], a_dst[n]=z[n].Wa[128:256]
// One wave per node, wave32 shuffle reduction. Also initializes m / denom.
// ---------------------------------------------------------------------------
__global__ __launch_bounds__(256)
void gat_node_prep(const float* __restrict__ z,
                   const float* __restrict__ Wa_l,
                   float* __restrict__ a_src,
                   float* __restrict__ a_dst,
                   float* __restrict__ mmax,
                   float* __restrict__ denom)
{
    int gid  = blockIdx.x * blockDim.x + threadIdx.x;
    int node = gid >> 5;
    int lane = gid & 31;
    if (node >= NNODE) return;

    const float4 zv = reinterpret_cast<const float4*>(z + (size_t)node * DD)[lane];
    const float* wa0 = Wa_l + lane * 4;        // Wa slice may be only 4B aligned
    const float* wa1 = Wa_l + DD + lane * 4;
    float s0 = zv.x * wa0[0] + zv.y * wa0[1] + zv.z * wa0[2] + zv.w * wa0[3];
    float s1 = zv.x * wa1[0] + zv.y * wa1[1] + zv.z * wa1[2] + zv.w * wa1[3];
    #pragma unroll
    for (int off = 16; off > 0; off >>= 1) {
        s0 += __shfl_xor(s0, off, 32);
        s1 += __shfl_xor(s1, off, 32);
    }
    if (lane == 0) {
        a_src[node] = s0;
        a_dst[node] = s1;
        mmax[node]  = -INFINITY;
        denom[node] = 0.0f;
    }
}

__device__ inline void atomicMaxF(float* addr, float val) {
    // Sign-aware monotone mapping of IEEE f32 onto integer compares.
    if (val >= 0.0f) atomicMax((int*)addr, __float_as_int(val));
    else             atomicMin((unsigned int*)addr, (unsigned int)__float_as_int(val));
}

// ---------------------------------------------------------------------------
// Edge pass 1: e = leaky_relu(a_src[s] + a_dst[t] + d*W0*Wa[256]); segment max
// ---------------------------------------------------------------------------
__global__ __launch_bounds__(256)
void gat_edge_logits(const int* __restrict__ src,
                     const int* __restrict__ dst,
                     const float* __restrict__ dfeat,
                     const float* __restrict__ W0_l,
                     const float* __restrict__ Wa_l,
                     const float* __restrict__ a_src,
                     const float* __restrict__ a_dst,
                     float* __restrict__ e,
                     float* __restrict__ mmax)
{
    int i = blockIdx.x * blockDim.x + threadIdx.x;
    if (i >= NEDGE) return;
    float ct = W0_l[0] * Wa_l[2 * DD];
    int s = src[i], t = dst[i];
    float v = a_src[s] + a_dst[t] + dfeat[i] * ct;
    v = (v > 0.0f) ? v : 0.01f * v;            // leaky_relu, slope 0.01
    e[i] = v;
    atomicMaxF(&mmax[t], v);
}

// ---------------------------------------------------------------------------
// Edge pass 2: ex = exp(e - m[dst]); segment sum into denom. e overwritten.
// ---------------------------------------------------------------------------
__global__ __launch_bounds__(256)
void gat_edge_exp(const int* __restrict__ dst,
                  float* __restrict__ e,
                  const float* __restrict__ mmax,
                  float* __restrict__ denom)
{
    int i = blockIdx.x * blockDim.x + threadIdx.x;
    if (i >= NEDGE) return;
    int t = dst[i];
    float x = __expf(e[i] - mmax[t]);
    e[i] = x;
    atomicAdd(&denom[t], x);
}

// ---------------------------------------------------------------------------
// Edge pass 3: znb[dst] += z[src] * (ex / denom[dst]).  One wave per edge,
// float4 per lane (32*4 = 128), f32 global atomics.
// ---------------------------------------------------------------------------
__global__ __launch_bounds__(256)
void gat_edge_scatter(const int* __restrict__ src,
                      const int* __restrict__ dst,
                      const float* __restrict__ ex,
                      const float* __restrict__ denom,
                      const float* __restrict__ z,
                      float* __restrict__ znb)
{
    int gid  = blockIdx.x * blockDim.x + threadIdx.x;
    int eidx = gid >> 5;
    int lane = gid & 31;
    if (eidx >= NEDGE) return;
    int s = src[eidx], t = dst[eidx];
    float alpha = ex[eidx] / denom[t];
    const float4 v = reinterpret_cast<const float4*>(z + (size_t)s * DD)[lane];
    float* o = znb + (size_t)t * DD + lane * 4;
    atomicAdd(o + 0, v.x * alpha);
    atomicAdd(o + 1, v.y * alpha);
    atomicAdd(o + 2, v.z * alpha);
    atomicAdd(o + 3, v.w * alpha);
}

// ---------------------------------------------------------------------------
// h_out = relu(z_i + z_neighbor)  (both already accumulated in znb)
// ---------------------------------------------------------------------------
__global__ __launch_bounds__(256)
void gat_relu(const float* __restrict__ znb, float* __restrict__ hout)
{
    int i = blockIdx.x * blockDim.x + threadIdx.x;
    if (i < NNODE * DD) hout[i] = fmaxf(znb[i], 0.0f);
}

extern "C" void kernel_launch(void* const* d_in, const int* in_sizes, int n_in,
                              void* d_out, int out_size, void* d_ws, size_t ws_size,
                              hipStream_t stream) {
    const float* attr = (const float*)d_in[0];   // [N, D]
    const float* dfe  = (const float*)d_in[1];   // [E, 1]
    const int*   src  = (const int*)d_in[2];     // [E]
    const int*   dst  = (const int*)d_in[3];     // [E]
    const float* W0   = (const float*)d_in[4];   // [L, 1, 1]
    const float* W1   = (const float*)d_in[5];   // [L, D, D]
    const float* W2   = (const float*)d_in[6];   // [L, D, D]
    const float* Wa   = (const float*)d_in[7];   // [L, 2D+1, 1]

    float* ws    = (float*)d_ws;
    float* hA    = ws;                               // N*D
    float* hB    = hA    + (size_t)NNODE * DD;       // N*D
    float* z     = hB    + (size_t)NNODE * DD;       // N*D
    float* znb   = z     + (size_t)NNODE * DD;       // N*D (z_i + aggregation)
    float* a_src = znb   + (size_t)NNODE * DD;       // N
    float* a_dst = a_src + NNODE;                    // N
    float* mmax  = a_dst + NNODE;                    // N
    float* denom = mmax  + NNODE;                    // N
    float* ebuf  = denom + NNODE;                    // E

    const int gemmBlocks    = NNODE / 16;                    // 2500
    const int nodeBlocks    = (NNODE * 32 + 255) / 256;      // 5000
    const int edgeBlocks    = (NEDGE + 255) / 256;           // 2500
    const int scatterBlocks = (NEDGE * 32 + 255) / 256;      // 80000
    const int reluBlocks    = (NNODE * DD + 255) / 256;      // 20000

    for (int l = 0; l < 3; ++l) {
        const float* hin  = (l == 0) ? attr : ((l == 1) ? hA : hB);
        float*       hout = (l == 2) ? (float*)d_out : ((l == 0) ? hA : hB);
        const float* W1l = W1 + (size_t)l * DD * DD;
        const float* W2l = W2 + (size_t)l * DD * DD;
        const float* W0l = W0 + l;
        const float* Wal = Wa + (size_t)l * (2 * DD + 1);

        gat_gemm_wmma<<<gemmBlocks, 256, 0, stream>>>(hin, W1l, W2l, z, znb);
        gat_node_prep<<<nodeBlocks, 256, 0, stream>>>(z, Wal, a_src, a_dst, mmax, denom);
        gat_edge_logits<<<edgeBlocks, 256, 0, stream>>>(src, dst, dfe, W0l, Wal,
                                                        a_src, a_dst, ebuf, mmax);
        gat_edge_exp<<<edgeBlocks, 256, 0, stream>>>(dst, ebuf, mmax, denom);
        gat_edge_scatter<<<scatterBlocks, 256, 0, stream>>>(src, dst, ebuf, denom, z, znb);
        gat_relu<<<reluBlocks, 256, 0, stream>>>(znb, hout);
    }
}